// Model_33294586479287
// MI455X (gfx1250) — compile-verified
//
#include <hip/hip_runtime.h>
#include <cstddef>

// ---------------------------------------------------------------------------
// DA-RNN forward for MI455X (gfx1250, wave32).
// - All GEMMs on V_WMMA_F32_16X16X4_F32 (f32 matrix pipe; recurrence needs f32).
// - input_encoded / enc_proj stored fp16 -> 134 MB total, resident in the
//   192 MB L2 across the decoder's 255 re-reads (instead of 68 GB from HBM).
// - Decoder split per-step: full-grid score kernel + small update kernel.
// - tanh via hardware v_exp_f32: tanh(x) = 1 - 2/(exp(2x)+1).
// ---------------------------------------------------------------------------

typedef __attribute__((ext_vector_type(2))) float    v2f;
typedef __attribute__((ext_vector_type(8))) float    v8f;
typedef __attribute__((ext_vector_type(8))) _Float16 v8h;

#define B_  512
#define T_  256
#define M_  81
#define MP_ 96      // M padded to multiple of 4 (zero fill) for WMMA K dim
#define HE_ 256
#define HD_ 256
#define BT  16      // batch rows per tile

static __device__ __forceinline__ float sigm(float x) {
  return 1.0f / (1.0f + __expf(-x));
}
// tanh via v_exp_f32; saturates correctly to +/-1 for large |x|.
static __device__ __forceinline__ float tanhfast(float x) {
  float e = __expf(2.0f * x);
  return 1.0f - 2.0f / (e + 1.0f);
}

// D = A(16x4) * B(4x16) + C, f32 WMMA.
static __device__ __forceinline__ v8f wmma4(v2f a, v2f b, v8f c) {
  return __builtin_amdgcn_wmma_f32_16x16x4_f32(false, a, false, b,
                                               (short)0, c, false, false);
}

// ---------------------------------------------------------------------------
// Kernel 1: attn_in[b][m] = sum_t x[b,t,m] * W_ai[t] + b_ai
// ---------------------------------------------------------------------------
__global__ __launch_bounds__(256) void attnin_kernel(
    const float* __restrict__ x, const float* __restrict__ W_ai,
    const float* __restrict__ b_ai, float* __restrict__ attn_in) {
  int idx = blockIdx.x * 256 + threadIdx.x;
  if (idx >= B_ * M_) return;
  int b = idx / M_, m = idx % M_;
  const float* xr = x + (size_t)b * T_ * M_ + m;
  float s = b_ai[0];
  for (int t = 0; t < T_; ++t) s += xr[(size_t)t * M_] * W_ai[t];
  attn_in[idx] = s;
}

// ---------------------------------------------------------------------------
// Kernel 2: encoder. One block = 16 batch rows, full T-step recurrence.
// 1024 threads = 32 wave32s (fills the WGP's 4 SIMD32s 8-deep for tanh work).
// ---------------------------------------------------------------------------
__global__ __launch_bounds__(1024) void encoder_kernel(
    const float* __restrict__ x,     const float* __restrict__ W_ah,
    const float* __restrict__ b_ah,  const float* __restrict__ W_a,
    const float* __restrict__ b_a,   const float* __restrict__ Wih_e,
    const float* __restrict__ Whh_e, const float* __restrict__ bih_e,
    const float* __restrict__ bhh_e, const float* __restrict__ attn_in,
    _Float16* __restrict__ enc_out /* (B,T,HE) fp16 */) {
  __shared__ float hc[BT][2 * HE_];    // [h | c]          32 KB (f32 state)
  __shared__ float hp[BT][T_];         // hid_proj         16 KB
  __shared__ float aw[BT][MP_];        // score -> softmax
  __shared__ float wi[BT][MP_];        // weighted input (zero-padded K)
  __shared__ float ain_s[BT][MP_];
  __shared__ float gbuf[BT][4 * HE_];  // gates            64 KB
  __shared__ float Wa_s[T_];

  const int tid  = threadIdx.x;
  const int lane = tid & 31;
  const int wave = tid >> 5;           // 0..31
  const int b0   = blockIdx.x * BT;
  const int row  = lane & 15;          // A row / B col / D col
  const int kb   = (lane >> 4) * 2;    // K sub-pair within K=4 slab
  const int m0   = (lane >> 4) * 8;    // D row base

  for (int i = tid; i < BT * 2 * HE_; i += 1024) ((float*)hc)[i] = 0.0f;
  if (tid < T_) Wa_s[tid] = W_a[tid];
  for (int i = tid; i < BT * MP_; i += 1024) {
    int b = i / MP_, m = i % MP_;
    ain_s[b][m] = (m < M_) ? attn_in[(b0 + b) * M_ + m] : 0.0f;
    wi[b][m]    = 0.0f;
  }
  __syncthreads();

  for (int t = 0; t < T_; ++t) {
    // ---- hid_proj = hc @ W_ah.T + b_ah : (16 x 256), K=512 --------------
    for (int tc = wave; tc < 16; tc += 32) {
      const int n0 = tc * 16;
      v8f acc = {};
      for (int k0 = 0; k0 < 2 * HE_; k0 += 4) {
        v2f a;  a.x = hc[row][k0 + kb];     a.y = hc[row][k0 + kb + 1];
        const float* wr = W_ah + (n0 + row) * (2 * HE_) + k0 + kb;
        v2f bb; bb.x = wr[0];               bb.y = wr[1];
        acc = wmma4(a, bb, acc);
      }
      const float bias = b_ah[n0 + row];
#pragma unroll
      for (int i = 0; i < 8; ++i) hp[m0 + i][n0 + row] = acc[i] + bias;
    }
    __syncthreads();

    // ---- score[b][m] = b_a + sum_t' tanh(hp[b][t'] + ain[b][m]) * W_a[t']
    for (int p = tid; p < BT * M_; p += 1024) {
      int b = p / M_, m = p % M_;
      const float ain = ain_s[b][m];
      float s = b_a[0];
      for (int tt = 0; tt < T_; ++tt)
        s += tanhfast(hp[b][tt] + ain) * Wa_s[tt];
      aw[b][m] = s;
    }
    __syncthreads();

    // ---- softmax over m, then wi = aw * x[:,t,:] ------------------------
    if (tid < BT) {
      const int b = tid;
      float mx = -1e30f;
      for (int m = 0; m < M_; ++m) mx = fmaxf(mx, aw[b][m]);
      float sum = 0.0f;
      for (int m = 0; m < M_; ++m) { float e = __expf(aw[b][m] - mx); aw[b][m] = e; sum += e; }
      const float inv = 1.0f / sum;
      const float* xr = x + ((size_t)(b0 + b) * T_ + t) * M_;
      for (int m = 0; m < M_; ++m) wi[b][m] = aw[b][m] * inv * xr[m];
    }
    __syncthreads();

    // ---- gates g = wi @ Wih_e.T + h @ Whh_e.T + biases : (16 x 1024) ----
    for (int tc = wave; tc < 64; tc += 32) {
      const int n0 = tc * 16;
      const int n  = n0 + row;
      v8f acc = {};
      for (int k0 = 0; k0 < MP_; k0 += 4) {          // K = 96 (81 padded)
        v2f a;  a.x = wi[row][k0 + kb];  a.y = wi[row][k0 + kb + 1];
        const int k = k0 + kb;
        v2f bb;
        bb.x = (k     < M_) ? Wih_e[n * M_ + k    ] : 0.0f;
        bb.y = (k + 1 < M_) ? Wih_e[n * M_ + k + 1] : 0.0f;
        acc = wmma4(a, bb, acc);
      }
      for (int k0 = 0; k0 < HE_; k0 += 4) {          // K = 256 (h part)
        v2f a;  a.x = hc[row][k0 + kb];  a.y = hc[row][k0 + kb + 1];
        const float* wr = Whh_e + n * HE_ + k0 + kb;
        v2f bb; bb.x = wr[0];            bb.y = wr[1];
        acc = wmma4(a, bb, acc);
      }
      const float bias = bih_e[n] + bhh_e[n];
#pragma unroll
      for (int i = 0; i < 8; ++i) gbuf[m0 + i][n] = acc[i] + bias;
    }
    __syncthreads();

    // ---- LSTM gate fusion; stream h (fp16) to enc_out -------------------
    for (int p = tid; p < BT * HE_; p += 1024) {
      const int b = p >> 8, e = p & 255;
      const float ig = gbuf[b][e],           fg = gbuf[b][HE_ + e];
      const float gg = gbuf[b][2 * HE_ + e], og = gbuf[b][3 * HE_ + e];
      const float c  = hc[b][HE_ + e];
      const float cn = sigm(fg) * c + sigm(ig) * tanhfast(gg);
      const float hn = sigm(og) * tanhfast(cn);
      hc[b][e] = hn;  hc[b][HE_ + e] = cn;
      enc_out[((size_t)(b0 + b) * T_ + t) * HE_ + e] = (_Float16)hn;
    }
    __syncthreads();
  }
}

// ---------------------------------------------------------------------------
// Kernel 3: enc_proj[(b,t)][h] = sum_e enc[(b,t)][e] * W_d1[h][512+e]  (fp16 out)
// ---------------------------------------------------------------------------
__global__ __launch_bounds__(256) void encproj_kernel(
    const _Float16* __restrict__ enc, const float* __restrict__ W_d1,
    _Float16* __restrict__ encp) {
  const int tid  = threadIdx.x;
  const int lane = tid & 31;
  const int wave = tid >> 5;
  const int r0   = blockIdx.x * 16;        // 16 (b,t) rows per block
  const int row  = lane & 15;
  const int kb   = (lane >> 4) * 2;
  const int m0   = (lane >> 4) * 8;

  for (int tc = wave * 2; tc < wave * 2 + 2; ++tc) {
    const int n0 = tc * 16;
    v8f acc = {};
    const _Float16* ar = enc + (size_t)(r0 + row) * HE_;
    for (int k0 = 0; k0 < HE_; k0 += 4) {
      v2f a;  a.x = (float)ar[k0 + kb];  a.y = (float)ar[k0 + kb + 1];
      const float* wr = W_d1 + (n0 + row) * (2 * HD_ + HE_) + 2 * HD_ + k0 + kb;
      v2f bb; bb.x = wr[0];              bb.y = wr[1];
      acc = wmma4(a, bb, acc);
    }
#pragma unroll
    for (int i = 0; i < 8; ++i)
      encp[(size_t)(r0 + m0 + i) * HE_ + n0 + row] = (_Float16)acc[i];
  }
}

// ---------------------------------------------------------------------------
// Kernel 4: decoder state init: hc = 0, hcproj = b_d1 (since hc==0)
// ---------------------------------------------------------------------------
__global__ __launch_bounds__(256) void dec_init_kernel(
    const float* __restrict__ b_d1, float* __restrict__ hc_g,
    float* __restrict__ hcproj) {
  int idx = blockIdx.x * 256 + threadIdx.x;
  if (idx < B_ * 2 * HD_) hc_g[idx] = 0.0f;
  if (idx < B_ * HE_)     hcproj[idx] = b_d1[idx & 255];
}

// ---------------------------------------------------------------------------
// Kernel 5 (per step): score[b][t] = b_d2 + sum_h tanh(encp + hcproj)*W_d2.
// Full-grid: 512 blocks (32 batch tiles x 16 t tiles) -> whole chip does tanh.
// encp is fp16 and L2-resident; 128-bit v8h loads + prefetch.
// ---------------------------------------------------------------------------
__global__ __launch_bounds__(256) void dec_score_kernel(
    const _Float16* __restrict__ encp, const float* __restrict__ hcproj,
    const float* __restrict__ W_d2, const float* __restrict__ b_d2,
    float* __restrict__ score) {
  __shared__ float hp_s[BT][HE_];
  __shared__ float Wd2_s[HE_];
  const int tid = threadIdx.x;
  const int b0  = (blockIdx.x >> 4) * BT;
  const int t0  = (blockIdx.x & 15) * 16;

  if (tid < HE_) Wd2_s[tid] = W_d2[tid];
  for (int i = tid; i < BT * HE_; i += 256)
    hp_s[i >> 8][i & 255] = hcproj[(b0 + (i >> 8)) * HE_ + (i & 255)];
  __syncthreads();

  const int b  = tid >> 4;
  const int tt = t0 + (tid & 15);
  const _Float16* er = encp + ((size_t)(b0 + b) * T_ + tt) * HE_;
  __builtin_prefetch(er, 0, 3);
  __builtin_prefetch(er + 128, 0, 3);
  float s = b_d2[0];
  for (int h8 = 0; h8 < HE_ / 8; ++h8) {
    v8h ch = *(const v8h*)(er + h8 * 8);
#pragma unroll
    for (int i = 0; i < 8; ++i) {
      const int h = h8 * 8 + i;
      s += tanhfast((float)ch[i] + hp_s[b][h]) * Wd2_s[h];
    }
  }
  score[(b0 + b) * T_ + tt] = s;
}

// ---------------------------------------------------------------------------
// Kernel 6 (per step): softmax, ctx, y_tild, decoder LSTM (WMMA), then
// pre-compute next step's hc_proj = hc @ W_d1[:, :512].T + b_d1 (WMMA).
// Emits final output on the last step.
// ---------------------------------------------------------------------------
__global__ __launch_bounds__(512) void dec_update_kernel(
    const _Float16* __restrict__ enc, const float* __restrict__ score,
    const float* __restrict__ y_history,
    const float* __restrict__ W_d1,  const float* __restrict__ b_d1,
    const float* __restrict__ W_fc,  const float* __restrict__ b_fc,
    const float* __restrict__ Wih_d, const float* __restrict__ Whh_d,
    const float* __restrict__ bih_d, const float* __restrict__ bhh_d,
    const float* __restrict__ W_ff,  const float* __restrict__ b_ff,
    float* __restrict__ hc_g, float* __restrict__ hcproj,
    float* __restrict__ out, int step) {
  __shared__ float hc[BT][2 * HD_];    // 32 KB
  __shared__ float aw[BT][T_];         // 16 KB
  __shared__ float ctx_s[BT][HE_];     // 16 KB
  __shared__ float gbuf[BT][4 * HD_];  // 64 KB
  __shared__ float yt_s[BT];

  const int tid  = threadIdx.x;
  const int lane = tid & 31;
  const int wave = tid >> 5;           // 0..15
  const int b0   = blockIdx.x * BT;
  const int row  = lane & 15;
  const int kb   = (lane >> 4) * 2;
  const int m0   = (lane >> 4) * 8;

  for (int i = tid; i < BT * 2 * HD_; i += 512)
    ((float*)hc)[i] = hc_g[b0 * 2 * HD_ + i];
  for (int i = tid; i < BT * T_; i += 512)
    aw[i >> 8][i & 255] = score[(b0 + (i >> 8)) * T_ + (i & 255)];
  __syncthreads();

  // ---- softmax over t per row -------------------------------------------
  if (tid < BT) {
    const int b = tid;
    float mx = -1e30f;
    for (int tt = 0; tt < T_; ++tt) mx = fmaxf(mx, aw[b][tt]);
    float sum = 0.0f;
    for (int tt = 0; tt < T_; ++tt) { float e = __expf(aw[b][tt] - mx); aw[b][tt] = e; sum += e; }
    const float inv = 1.0f / sum;
    for (int tt = 0; tt < T_; ++tt) aw[b][tt] *= inv;
  }
  __syncthreads();

  // ---- ctx[b][e] = sum_t aw[b][t] * enc[b,t,e] (fp16, v8h loads) --------
  {
    const int e8 = (tid & 31) * 8;     // 32 chunks of 8 e-values
    const int b  = tid >> 5;           // 16 rows
    float acc[8];
#pragma unroll
    for (int i = 0; i < 8; ++i) acc[i] = 0.0f;
    const _Float16* er = enc + (size_t)(b0 + b) * T_ * HE_ + e8;
    for (int tt = 0; tt < T_; ++tt) {
      const float a = aw[b][tt];
      v8h ch = *(const v8h*)(er + (size_t)tt * HE_);
#pragma unroll
      for (int i = 0; i < 8; ++i) acc[i] += a * (float)ch[i];
    }
#pragma unroll
    for (int i = 0; i < 8; ++i) ctx_s[b][e8 + i] = acc[i];
  }
  __syncthreads();

  // ---- y_tild[b] = [ctx, y_t] @ W_fc.T + b_fc ---------------------------
  if (tid < BT) {
    const int b = tid;
    float s = b_fc[0] + y_history[(size_t)(b0 + b) * (T_ - 1) + step] * W_fc[HE_];
    for (int e = 0; e < HE_; ++e) s += ctx_s[b][e] * W_fc[e];
    yt_s[b] = s;
  }
  __syncthreads();

  // ---- gates = y_tild*Wih_d + h @ Whh_d.T + biases : (16 x 1024) --------
  for (int tc = wave; tc < 64; tc += 16) {
    const int n0 = tc * 16;
    const int n  = n0 + row;
    v8f acc = {};
    for (int k0 = 0; k0 < HD_; k0 += 4) {
      v2f a;  a.x = hc[row][k0 + kb];  a.y = hc[row][k0 + kb + 1];
      const float* wr = Whh_d + n * HD_ + k0 + kb;
      v2f bb; bb.x = wr[0];            bb.y = wr[1];
      acc = wmma4(a, bb, acc);
    }
    const float bias = bih_d[n] + bhh_d[n];
    const float wih  = Wih_d[n];
#pragma unroll
    for (int i = 0; i < 8; ++i)
      gbuf[m0 + i][n] = acc[i] + bias + yt_s[m0 + i] * wih;
  }
  __syncthreads();

  // ---- LSTM gate fusion; write hc back to global ------------------------
  for (int p = tid; p < BT * HD_; p += 512) {
    const int b = p >> 8, e = p & 255;
    const float ig = gbuf[b][e],           fg = gbuf[b][HD_ + e];
    const float gg = gbuf[b][2 * HD_ + e], og = gbuf[b][3 * HD_ + e];
    const float c  = hc[b][HD_ + e];
    const float cn = sigm(fg) * c + sigm(ig) * tanhfast(gg);
    const float hn = sigm(og) * tanhfast(cn);
    hc[b][e] = hn;  hc[b][HD_ + e] = cn;
    hc_g[(b0 + b) * 2 * HD_ + e]       = hn;
    hc_g[(b0 + b) * 2 * HD_ + HD_ + e] = cn;
  }
  __syncthreads();

  if (step == T_ - 2) {
    // ---- out[b] = [h_fin, ctx_last] @ W_ff.T + b_ff ---------------------
    if (tid < BT) {
      const int b = tid;
      float s = b_ff[0];
      for (int e = 0; e < HE_; ++e)
        s += hc[b][e] * W_ff[e] + ctx_s[b][e] * W_ff[HE_ + e];
      out[b0 + b] = s;
    }
  } else {
    // ---- next hc_proj = hc @ W_d1[:, :512].T + b_d1 (WMMA, K=512) -------
    for (int tc = wave; tc < 16; tc += 16) {
      const int n0 = tc * 16;
      v8f acc = {};
      for (int k0 = 0; k0 < 2 * HD_; k0 += 4) {
        v2f a;  a.x = hc[row][k0 + kb];  a.y = hc[row][k0 + kb + 1];
        const float* wr = W_d1 + (n0 + row) * (2 * HD_ + HE_) + k0 + kb;
        v2f bb; bb.x = wr[0];            bb.y = wr[1];
        acc = wmma4(a, bb, acc);
      }
      const float bias = b_d1[n0 + row];
#pragma unroll
      for (int i = 0; i < 8; ++i)
        hcproj[(b0 + m0 + i) * HE_ + n0 + row] = acc[i] + bias;
    }
  }
}

// ---------------------------------------------------------------------------
extern "C" void kernel_launch(void* const* d_in, const int* in_sizes, int n_in,
                              void* d_out, int out_size, void* d_ws, size_t ws_size,
                              hipStream_t stream) {
  const float* x         = (const float*)d_in[0];
  const float* y_history = (const float*)d_in[1];
  const float* W_ah      = (const float*)d_in[2];
  const float* b_ah      = (const float*)d_in[3];
  const float* W_ai      = (const float*)d_in[4];
  const float* b_ai      = (const float*)d_in[5];
  const float* W_a       = (const float*)d_in[6];
  const float* b_a       = (const float*)d_in[7];
  const float* Wih_e     = (const float*)d_in[8];
  const float* Whh_e     = (const float*)d_in[9];
  const float* bih_e     = (const float*)d_in[10];
  const float* bhh_e     = (const float*)d_in[11];
  const float* W_d1      = (const float*)d_in[12];
  const float* b_d1      = (const float*)d_in[13];
  const float* W_d2      = (const float*)d_in[14];
  const float* b_d2      = (const float*)d_in[15];
  const float* W_fc      = (const float*)d_in[16];
  const float* b_fc      = (const float*)d_in[17];
  const float* Wih_d     = (const float*)d_in[18];
  const float* Whh_d     = (const float*)d_in[19];
  const float* bih_d     = (const float*)d_in[20];
  const float* bhh_d     = (const float*)d_in[21];
  const float* W_ff      = (const float*)d_in[22];
  const float* b_ff      = (const float*)d_in[23];
  float* out = (float*)d_out;

  // Workspace layout (bytes)
  const size_t enc_elems = (size_t)B_ * T_ * HE_;           // 33,554,432
  char* ws = (char*)d_ws;
  _Float16* ws_enc  = (_Float16*)ws;                        // 67 MB fp16
  _Float16* ws_encp = (_Float16*)(ws + enc_elems * 2);      // 67 MB fp16
  float* ws_ain     = (float*)(ws + enc_elems * 4);         // B*M f32
  float* ws_hc      = ws_ain + (size_t)B_ * M_;             // B*512 f32
  float* ws_hcproj  = ws_hc + (size_t)B_ * 2 * HD_;         // B*256 f32
  float* ws_score   = ws_hcproj + (size_t)B_ * HE_;         // B*256 f32

  attnin_kernel<<<(B_ * M_ + 255) / 256, 256, 0, stream>>>(x, W_ai, b_ai, ws_ain);

  encoder_kernel<<<B_ / BT, 1024, 0, stream>>>(
      x, W_ah, b_ah, W_a, b_a, Wih_e, Whh_e, bih_e, bhh_e, ws_ain, ws_enc);

  encproj_kernel<<<(B_ * T_) / 16, 256, 0, stream>>>(ws_enc, W_d1, ws_encp);

  dec_init_kernel<<<(B_ * 2 * HD_ + 255) / 256, 256, 0, stream>>>(b_d1, ws_hc, ws_hcproj);

  for (int step = 0; step < T_ - 1; ++step) {
    dec_score_kernel<<<(B_ / BT) * 16, 256, 0, stream>>>(
        ws_encp, ws_hcproj, W_d2, b_d2, ws_score);
    dec_update_kernel<<<B_ / BT, 512, 0, stream>>>(
        ws_enc, ws_score, y_history, W_d1, b_d1, W_fc, b_fc, Wih_d, Whh_d,
        bih_d, bhh_d, W_ff, b_ff, ws_hc, ws_hcproj, out, step);
  }
}